// mv_embedding_13022340841686
// MI455X (gfx1250) — compile-verified
//
#include <hip/hip_runtime.h>
#include <hip/hip_bf16.h>

// Problem sizes (match reference)
#define B_SZ   512
#define K_SZ   256
#define L_SZ   1000
#define E_SZ   3
#define OUT_SZ 128
#define HID    (E_SZ * K_SZ)   // 768

typedef float v2f __attribute__((ext_vector_type(2)));
typedef float v8f __attribute__((ext_vector_type(8)));

// ---------------------------------------------------------------------------
// Stage 1: y[b,k,e] = sum_l x[b,k,l] * W_emb[e,l]
// Memory-bound streaming kernel: one wave32 per (b,k) row of 1000 floats.
// W_emb (3x1000 f32 = 12 KB) is staged in LDS once per workgroup.
// ---------------------------------------------------------------------------
__global__ __launch_bounds__(256) void embed_kernel(const float* __restrict__ x,
                                                    const float* __restrict__ Wemb,
                                                    float* __restrict__ y)
{
    __shared__ __align__(16) float sW[E_SZ * L_SZ];   // 3000 floats = 12 KB

    const int tid = threadIdx.x;
    for (int i = tid; i < E_SZ * L_SZ; i += 256) sW[i] = Wemb[i];
    __syncthreads();

    const int wave = tid >> 5;
    const int lane = tid & 31;
    const long row = (long)blockIdx.x * 8 + wave;     // 0 .. B*K-1
    const float* __restrict__ xr = x + row * L_SZ;

    float a0 = 0.f, a1 = 0.f, a2 = 0.f;

    // 7 full iterations: lanes cover 128 floats per iteration (float4 each)
    #pragma unroll
    for (int it = 0; it < 7; ++it) {
        const int l = it * 128 + lane * 4;
        const float4 xv = *(const float4*)(xr + l);
        const float4 w0 = *(const float4*)(sW + l);
        const float4 w1 = *(const float4*)(sW + L_SZ + l);
        const float4 w2 = *(const float4*)(sW + 2 * L_SZ + l);
        a0 += xv.x * w0.x + xv.y * w0.y + xv.z * w0.z + xv.w * w0.w;
        a1 += xv.x * w1.x + xv.y * w1.y + xv.z * w1.z + xv.w * w1.w;
        a2 += xv.x * w2.x + xv.y * w2.y + xv.z * w2.z + xv.w * w2.w;
    }
    // tail: elements 896..999 (104 floats = 26 lanes * float4)
    {
        const int l = 896 + lane * 4;
        if (l < L_SZ) {
            const float4 xv = *(const float4*)(xr + l);
            const float4 w0 = *(const float4*)(sW + l);
            const float4 w1 = *(const float4*)(sW + L_SZ + l);
            const float4 w2 = *(const float4*)(sW + 2 * L_SZ + l);
            a0 += xv.x * w0.x + xv.y * w0.y + xv.z * w0.z + xv.w * w0.w;
            a1 += xv.x * w1.x + xv.y * w1.y + xv.z * w1.z + xv.w * w1.w;
            a2 += xv.x * w2.x + xv.y * w2.y + xv.z * w2.z + xv.w * w2.w;
        }
    }

    // wave32 butterfly reduction
    #pragma unroll
    for (int m = 16; m >= 1; m >>= 1) {
        a0 += __shfl_xor(a0, m, 32);
        a1 += __shfl_xor(a1, m, 32);
        a2 += __shfl_xor(a2, m, 32);
    }

    if (lane == 0) {
        float* __restrict__ yo = y + row * E_SZ;   // kernel-major [B, K, 3] -> [B, 768]
        yo[0] = a0;
        yo[1] = a1;
        yo[2] = a2;
    }
}

// ---------------------------------------------------------------------------
// Stage 2: out[512,128] = y[512,768] @ W_fc2^T + b_fc2, full f32 via
// V_WMMA_F32_16X16X4_F32.  One workgroup per 16-row M strip; the 16x768 f32
// A strip (48 KB) is staged in LDS; each of the 8 waves owns one 16-wide N
// tile and streams B fragments (8B global loads, L2-resident).
//
// Fragment layouts (ISA 7.12.2, 32-bit data):
//   A 16x4 : VGPR0 = K=0 (lanes 0-15) / K=2 (lanes 16-31); VGPR1 = K=1 / K=3
//   B 4x16 : mirrored, N striped across lanes
//   C/D    : VGPR i -> M=i (lanes 0-15), M=i+8 (lanes 16-31), N=lane&15
// ---------------------------------------------------------------------------
__global__ __launch_bounds__(256) void fc2_wmma_kernel(const float* __restrict__ y,
                                                       const float* __restrict__ Wfc2,
                                                       const float* __restrict__ bias,
                                                       float* __restrict__ out)
{
    __shared__ __align__(16) float sA[16 * HID];      // 48 KB

    const int tid = threadIdx.x;
    const int m0  = blockIdx.x * 16;

    // stage A strip (contiguous 16*768 floats) into LDS, coalesced float4
    const float* __restrict__ ya = y + (long)m0 * HID;
    for (int i = tid * 4; i < 16 * HID; i += 256 * 4) {
        *(float4*)(sA + i) = *(const float4*)(ya + i);
    }
    __syncthreads();

    const int wave = tid >> 5;
    const int lane = tid & 31;
    const int n0   = wave * 16;
    const int half = lane >> 4;      // 0: K+{0,1}, 1: K+{2,3}
    const int idx  = lane & 15;      // M row (for A) / N col (for B)

    v8f c = {};                      // f32 accumulator, starts at 0

    const float* __restrict__ arow = sA + idx * HID + half * 2;
    const float* __restrict__ brow = Wfc2 + (long)(n0 + idx) * HID + half * 2;

    #pragma unroll 4
    for (int kk = 0; kk < HID / 4; ++kk) {           // 192 k-steps of 4
        const v2f a = *(const v2f*)(arow + kk * 4);
        const v2f b = *(const v2f*)(brow + kk * 4);
        c = __builtin_amdgcn_wmma_f32_16x16x4_f32(
                /*neg_a=*/false, a, /*neg_b=*/false, b,
                /*c_mod=*/(short)0, c, /*reuse_a=*/false, /*reuse_b=*/false);
    }

    const float bv = bias[n0 + idx];
    #pragma unroll
    for (int i = 0; i < 8; ++i) {
        const int row = m0 + i + half * 8;
        out[row * OUT_SZ + n0 + idx] = c[i] + bv;
    }
}

// ---------------------------------------------------------------------------
extern "C" void kernel_launch(void* const* d_in, const int* in_sizes, int n_in,
                              void* d_out, int out_size, void* d_ws, size_t ws_size,
                              hipStream_t stream)
{
    const float* x     = (const float*)d_in[0];   // [512, 256, 1000]
    const float* W_emb = (const float*)d_in[1];   // [3, 1000]
    const float* W_fc2 = (const float*)d_in[2];   // [128, 768]
    const float* b_fc2 = (const float*)d_in[3];   // [128]
    float* out = (float*)d_out;                   // [512, 128]
    float* y   = (float*)d_ws;                    // [512, 768] = 1.5 MB scratch

    (void)in_sizes; (void)n_in; (void)out_size; (void)ws_size;

    // Stage 1: 131072 rows, 8 waves (rows) per 256-thread block
    const int rows   = B_SZ * K_SZ;
    const int blocks = rows / 8;                  // 16384
    embed_kernel<<<blocks, 256, 0, stream>>>(x, W_emb, y);

    // Stage 2: one block per 16-row strip of the output
    fc2_wmma_kernel<<<B_SZ / 16, 256, 0, stream>>>(y, W_fc2, b_fc2, out);
}